// APENBlock_21947282883129
// MI455X (gfx1250) — compile-verified
//
#include <hip/hip_runtime.h>
#include <math.h>

typedef __attribute__((ext_vector_type(2))) float v2f;
typedef __attribute__((ext_vector_type(8))) float v8f;

#define BB   4
#define NN   2048
#define GG   8
#define KK   16
#define NOPS 8
#define THRW 0.1f

// activation LDS row stride (padded 96->98: conflict-free b64 A loads)
#define ACTSTR 98
#define ACTSZ  (16 * ACTSTR)

// ---- workspace layout (float offsets) ----
// fused weights stored K-pair interleaved: [k/2][cout][2] with padded row
// stride (cout*2 rounded so stride % 64 == 32) for bank-conflict-free b64.
// rows(cin/2): {26,16,24,32,40}, strides: {96,96,160,160,224}
#define OFF_WF 0          // 26*96+16*96+24*160+32*160+40*224 = 21952 floats
#define OFF_BF 21952      // fused bias, 320
#define OFF_FR 22272      // frames B*G*9 = 288
#define OFF_CT 22560      // centers 96
#define OFF_NM 22656      // nmean 96
#define OFF_PN 22752      // gathered neighbors B*N*K*3 = 393216
// total = 415968 floats ~= 1.66 MB

// ============================================================
// Kernel 0: fold BN into weights/bias; write K-pair-interleaved layout
// ============================================================
__global__ void prep_layer(const float* __restrict__ W, const float* __restrict__ bb,
                           const float* __restrict__ ss, const float* __restrict__ oo,
                           float* __restrict__ wf, float* __restrict__ bf,
                           int cin_real, int cin_pad, int cout, int wstr) {
  int i = blockIdx.x * blockDim.x + threadIdx.x;
  if (i < cin_pad * cout) {
    int r = i / cout, c = i % cout;
    float v = (r < cin_real) ? W[r * cout + c] * ss[c] : 0.f;
    wf[(r >> 1) * wstr + c * 2 + (r & 1)] = v;
  }
  if (i < cout) bf[i] = bb[i] * ss[i] + oo[i];
}

// ============================================================
// 3x3 symmetric eigendecomposition helpers
// ============================================================
__device__ inline void cross3(const float* a, const float* b, float* c) {
  c[0] = a[1] * b[2] - a[2] * b[1];
  c[1] = a[2] * b[0] - a[0] * b[2];
  c[2] = a[0] * b[1] - a[1] * b[0];
}
__device__ inline void normalize3(float* v) {
  float n = v[0] * v[0] + v[1] * v[1] + v[2] * v[2];
  float r = (n > 0.f) ? rsqrtf(n) : 0.f;
  v[0] *= r; v[1] *= r; v[2] *= r;
}
__device__ inline void signfix3(float* v) {
  float ax = fabsf(v[0]), ay = fabsf(v[1]), az = fabsf(v[2]);
  float m = (ax >= ay && ax >= az) ? v[0] : ((ay >= az) ? v[1] : v[2]);
  if (m < 0.f) { v[0] = -v[0]; v[1] = -v[1]; v[2] = -v[2]; }
}
__device__ void eigvec3(float a00, float a01, float a02, float a11, float a12, float a22,
                        float lam, float* v) {
  float r0[3] = {a00 - lam, a01, a02};
  float r1[3] = {a01, a11 - lam, a12};
  float r2[3] = {a02, a12, a22 - lam};
  float c0[3], c1[3], c2[3];
  cross3(r0, r1, c0); cross3(r0, r2, c1); cross3(r1, r2, c2);
  float n0 = c0[0]*c0[0] + c0[1]*c0[1] + c0[2]*c0[2];
  float n1 = c1[0]*c1[0] + c1[1]*c1[1] + c1[2]*c1[2];
  float n2 = c2[0]*c2[0] + c2[1]*c2[1] + c2[2]*c2[2];
  const float* best = c0; float nb = n0;
  if (n1 > nb) { best = c1; nb = n1; }
  if (n2 > nb) { best = c2; nb = n2; }
  v[0] = best[0]; v[1] = best[1]; v[2] = best[2];
  normalize3(v);
}

// ============================================================
// Kernel 1: per-(b,g) moments -> center, nmean, covariance, eigh
// ============================================================
__global__ void frames_kernel(const float* __restrict__ pc, const float* __restrict__ nrm,
                              const float* __restrict__ drw, float* __restrict__ ws) {
  __shared__ float acc[13];
  int tid = threadIdx.x;
  if (tid < 13) acc[tid] = 0.f;
  __syncthreads();
  int b = blockIdx.x / GG, g = blockIdx.x % GG;
  float s[13] = {0,0,0,0,0,0,0,0,0,0,0,0,0};
  for (int n = tid; n < NN; n += blockDim.x) {
    float wv = drw[(b * NN + n) * GG + g];
    const float* p = pc + (size_t)(b * NN + n) * 3;
    const float* q = nrm + (size_t)(b * NN + n) * 3;
    float x = p[0], y = p[1], z = p[2];
    s[0] += wv; s[1] += wv*x; s[2] += wv*y; s[3] += wv*z;
    s[4] += wv*x*x; s[5] += wv*x*y; s[6] += wv*x*z;
    s[7] += wv*y*y; s[8] += wv*y*z; s[9] += wv*z*z;
    s[10] += wv*q[0]; s[11] += wv*q[1]; s[12] += wv*q[2];
  }
  #pragma unroll
  for (int i = 0; i < 13; ++i) atomicAdd(&acc[i], s[i]);   // ds_add_f32
  __syncthreads();
  if (tid == 0) {
    float SW = acc[0];
    float inv = 1.f / (SW + 1e-6f);
    float cx = acc[1] * inv, cy = acc[2] * inv, cz = acc[3] * inv;
    float a00 = acc[4] - 2.f*cx*acc[1] + SW*cx*cx;
    float a01 = acc[5] - cx*acc[2] - cy*acc[1] + SW*cx*cy;
    float a02 = acc[6] - cx*acc[3] - cz*acc[1] + SW*cx*cz;
    float a11 = acc[7] - 2.f*cy*acc[2] + SW*cy*cy;
    float a12 = acc[8] - cy*acc[3] - cz*acc[2] + SW*cy*cz;
    float a22 = acc[9] - 2.f*cz*acc[3] + SW*cz*cz;

    float v0c[3], v1c[3], v2c[3];
    float p1 = a01*a01 + a02*a02 + a12*a12;
    if (p1 < 1e-18f) {
      int i0 = 0, i1 = 1, i2 = 2;
      float d0 = a00, d1 = a11, d2 = a22, t; int ti;
      if (d0 > d1) { t = d0; d0 = d1; d1 = t; ti = i0; i0 = i1; i1 = ti; }
      if (d1 > d2) { t = d1; d1 = d2; d2 = t; ti = i1; i1 = i2; i2 = ti; }
      if (d0 > d1) { t = d0; d0 = d1; d1 = t; ti = i0; i0 = i1; i1 = ti; }
      v0c[0] = (i0 == 0); v0c[1] = (i0 == 1); v0c[2] = (i0 == 2);
      v1c[0] = (i1 == 0); v1c[1] = (i1 == 1); v1c[2] = (i1 == 2);
      v2c[0] = (i2 == 0); v2c[1] = (i2 == 1); v2c[2] = (i2 == 2);
    } else {
      float q = (a00 + a11 + a22) / 3.f;
      float p2 = (a00-q)*(a00-q) + (a11-q)*(a11-q) + (a22-q)*(a22-q) + 2.f*p1;
      float p = sqrtf(p2 / 6.f);
      float ip = 1.f / p;
      float b00 = (a00-q)*ip, b01 = a01*ip, b02 = a02*ip;
      float b11 = (a11-q)*ip, b12 = a12*ip, b22 = (a22-q)*ip;
      float detB = b00*(b11*b22 - b12*b12) - b01*(b01*b22 - b12*b02) + b02*(b01*b12 - b11*b02);
      float r = fminf(1.f, fmaxf(-1.f, 0.5f * detB));
      float phi = acosf(r) / 3.f;
      float lmax = q + 2.f * p * cosf(phi);
      float lmin = q + 2.f * p * cosf(phi + 2.09439510239f);
      eigvec3(a00, a01, a02, a11, a12, a22, lmin, v0c);
      eigvec3(a00, a01, a02, a11, a12, a22, lmax, v2c);
      cross3(v2c, v0c, v1c); normalize3(v1c);
    }
    signfix3(v0c); signfix3(v1c); signfix3(v2c);

    float* fr = ws + OFF_FR + (b * GG + g) * 9;
    float* ct = ws + OFF_CT + (b * GG + g) * 3;
    float* nm = ws + OFF_NM + (b * GG + g) * 3;
    #pragma unroll
    for (int d = 0; d < 3; ++d) {
      fr[d * 3 + 0] = v0c[d];
      fr[d * 3 + 1] = v1c[d];
      fr[d * 3 + 2] = v2c[d];
      ct[d] = (d == 0) ? cx : (d == 1) ? cy : cz;
      nm[d] = acc[10 + d] * inv;
    }
  }
}

// ============================================================
// Kernel 2: soft-weighted kNN (top-16 per query), pre-gather coords
// ============================================================
__global__ void knn_kernel(const float* __restrict__ pc, const float* __restrict__ drw,
                           float* __restrict__ ws) {
  __shared__ float sp[128 * 3];
  __shared__ float swd[128 * 8];
  int tid = threadIdx.x;
  int tilesPerB = NN / 128;
  int b = blockIdx.x / tilesPerB;
  int n = (blockIdx.x % tilesPerB) * 128 + tid;

  float pq[3], wq[8];
  #pragma unroll
  for (int d = 0; d < 3; ++d) pq[d] = pc[(size_t)(b * NN + n) * 3 + d];
  #pragma unroll
  for (int g = 0; g < 8; ++g) wq[g] = drw[(size_t)(b * NN + n) * GG + g];

  float bd[KK]; int bi[KK];
  #pragma unroll
  for (int k = 0; k < KK; ++k) { bd[k] = 3.0e38f; bi[k] = 0; }

  for (int m0 = 0; m0 < NN; m0 += 128) {
    __syncthreads();
    #pragma unroll
    for (int d = 0; d < 3; ++d) sp[tid * 3 + d] = pc[(size_t)(b * NN + m0 + tid) * 3 + d];
    #pragma unroll
    for (int g = 0; g < 8; ++g) swd[tid * 8 + g] = drw[(size_t)(b * NN + m0 + tid) * GG + g];
    __syncthreads();
    for (int j = 0; j < 128; ++j) {
      float ww = 0.f;
      #pragma unroll
      for (int g = 0; g < 8; ++g) ww += wq[g] * swd[j * 8 + g];
      float dx = pq[0] - sp[j * 3 + 0];
      float dy = pq[1] - sp[j * 3 + 1];
      float dz = pq[2] - sp[j * 3 + 2];
      float sq = dx * dx + dy * dy + dz * dz;
      float dv = ww * (sq - 1000.f) + 1000.f;   // ww*sq + (1-ww)*1000
      if (dv < bd[KK - 1]) {
        float cd = dv; int ci = m0 + j;
        #pragma unroll
        for (int q = 0; q < KK; ++q) {          // stable insert (strict <)
          if (cd < bd[q]) {
            float td = bd[q]; int ti = bi[q];
            bd[q] = cd; bi[q] = ci; cd = td; ci = ti;
          }
        }
      }
    }
  }
  float* pn = ws + OFF_PN + (size_t)(b * NN + n) * KK * 3;
  #pragma unroll
  for (int k = 0; k < KK; ++k) {
    int m = bi[k];
    #pragma unroll
    for (int d = 0; d < 3; ++d) pn[k * 3 + d] = pc[(size_t)(b * NN + m) * 3 + d];
  }
}

// ============================================================
// Kernel 3: feature build + 5-layer MLP on the f32 WMMA pipe
// block = 128 thr (4 waves); each wave owns a 16-row M-tile.
// Weights async-staged to LDS (GLOBAL_LOAD_ASYNC_TO_LDS_B128).
// ============================================================
__global__ void mlp_kernel(const float* __restrict__ nrm, const float* __restrict__ drw,
                           const float* __restrict__ gwf, const float* __restrict__ gbf,
                           const float* __restrict__ gfr, const float* __restrict__ gct,
                           const float* __restrict__ gnm, const float* __restrict__ gpn,
                           float* __restrict__ out) {
  extern __shared__ float smem[];
  float* wbuf  = smem;            // 8960 max: layer weights, interleaved+padded
  float* bbuf  = wbuf + 8960;     // 96
  float* fbuf  = bbuf + 96;       // 9 frames
  float* cbuf  = fbuf + 9;        // 3 center
  float* mbuf  = cbuf + 3;        // 3 nmean
  float* wnbuf = mbuf + 3;        // 64 wn per row
  float* act   = wnbuf + 64;      // 4 waves * 2 bufs * 16 * ACTSTR

  int t = blockIdx.x;
  int ntile = t & 31; t >>= 5;
  int g = t & 7;      t >>= 3;
  int o = t & 7;      t >>= 3;
  int b = t;

  int tid  = threadIdx.x;
  int wv   = tid >> 5;
  int lane = tid & 31;
  int rloc = lane & 15;
  int hset = lane >> 4;
  int nbase = ntile * 64;
  int nwave = nbase + wv * 16;
  int rown  = nwave + rloc;

  if (tid < 9)              fbuf[tid]      = gfr[(b * GG + g) * 9 + tid];
  else if (tid < 12)        cbuf[tid - 9]  = gct[(b * GG + g) * 3 + (tid - 9)];
  else if (tid < 15)        mbuf[tid - 12] = gnm[(b * GG + g) * 3 + (tid - 12)];
  if (tid < 64) wnbuf[tid] = drw[((size_t)b * NN + nbase + tid) * GG + g];
  __syncthreads();

  // ---- feature build: x[row][0..51] into act buffer 0 of this wave ----
  float F[3][3];
  #pragma unroll
  for (int e = 0; e < 3; ++e) {
    float sgn = ((o >> (2 - e)) & 1) ? -1.f : 1.f;
    #pragma unroll
    for (int d = 0; d < 3; ++d) F[d][e] = fbuf[d * 3 + e] * sgn;
  }
  float wn = wnbuf[wv * 16 + rloc];
  float c0 = cbuf[0], c1 = cbuf[1], c2 = cbuf[2];
  float* aF = act + (wv * 2 + 0) * ACTSZ;
  {
    const float* pn = gpn + ((size_t)(b * NN + rown) * KK + hset * 8) * 3;
    float qv[24];
    #pragma unroll
    for (int i = 0; i < 24; ++i) qv[i] = pn[i];
    #pragma unroll
    for (int k = 0; k < 8; ++k) {
      float vx = wn * (qv[k * 3 + 0] - c0);
      float vy = wn * (qv[k * 3 + 1] - c1);
      float vz = wn * (qv[k * 3 + 2] - c2);
      #pragma unroll
      for (int e = 0; e < 3; ++e)
        aF[rloc * ACTSTR + hset * 24 + k * 3 + e] = vx * F[0][e] + vy * F[1][e] + vz * F[2][e];
    }
    if (hset == 1) {
      float nx = nrm[((size_t)b * NN + rown) * 3 + 0];
      float ny = nrm[((size_t)b * NN + rown) * 3 + 1];
      float nz = nrm[((size_t)b * NN + rown) * 3 + 2];
      float ux = wn * nx + (1.f - wn) * mbuf[0];
      float uy = wn * ny + (1.f - wn) * mbuf[1];
      float uz = wn * nz + (1.f - wn) * mbuf[2];
      #pragma unroll
      for (int e = 0; e < 3; ++e)
        aF[rloc * ACTSTR + 48 + e] = ux * F[0][e] + uy * F[1][e] + uz * F[2][e];
      aF[rloc * ACTSTR + 51] = 0.f;   // pad 51 -> 52
    }
  }

  // mask per D-fragment row (wn >= THR), once
  float msk[8];
  #pragma unroll
  for (int v = 0; v < 8; ++v)
    msk[v] = (wnbuf[wv * 16 + v + 8 * hset] >= THRW) ? 1.f : 0.f;

  // ---- 5-layer MLP: D = A(16xK) x B(Kx16) + C, fp32 WMMA, K-step 4 ----
  const int CINL[5]  = {52, 32, 48, 64, 80};
  const int COUTL[5] = {32, 48, 64, 80, 96};
  const int WROW[5]  = {26, 16, 24, 32, 40};        // cin/2
  const int WSTR[5]  = {96, 96, 160, 160, 224};     // padded interleaved stride
  const int OWL[5]   = {0, 2496, 4032, 7872, 12992};
  const int OBL[5]   = {0, 32, 80, 144, 224};
  int bsel = 0;
  int col = rloc;
  #pragma unroll
  for (int l = 0; l < 5; ++l) {
    __syncthreads();
    {
      // async DMA of the interleaved weight image into LDS (16B/lane chunks)
      const float* src = gwf + OWL[l];
      int nchunk = (WROW[l] * WSTR[l]) >> 2;
      for (int i = tid; i < nchunk; i += 128) {
        unsigned lds_off = (unsigned)(uintptr_t)(wbuf + (i << 2));
        unsigned long long ga = (unsigned long long)(uintptr_t)(src + (i << 2));
        asm volatile("global_load_async_to_lds_b128 %0, %1, off"
                     :: "v"(lds_off), "v"(ga) : "memory");
      }
      if (tid < COUTL[l]) bbuf[tid] = gbf[OBL[l] + tid];
      if (l < 4 && tid == 0) __builtin_prefetch(gwf + OWL[l + 1], 0, 1);
      asm volatile("s_wait_asynccnt 0" ::: "memory");
    }
    __syncthreads();

    const float* aI = act + (wv * 2 + bsel) * ACTSZ;
    float* aO = act + (wv * 2 + (bsel ^ 1)) * ACTSZ;
    int cin = CINL[l], cout = COUTL[l];
    for (int n0c = 0; n0c < cout; n0c += 16) {
      v8f accv;
      float bias = bbuf[n0c + col];
      #pragma unroll
      for (int v = 0; v < 8; ++v) accv[v] = bias;
      for (int k0 = 0; k0 < cin; k0 += 4) {
        // A fragment: rows = lane%16, K pair (k0+2*hset, +1)   -> ds_load_b64
        v2f Af = *(const v2f*)(aI + rloc * ACTSTR + k0 + 2 * hset);
        // B fragment: interleaved [k/2][col][2]                -> ds_load_b64
        v2f Bf = *(const v2f*)(wbuf + ((k0 >> 1) + hset) * WSTR[l] + (n0c + col) * 2);
        accv = __builtin_amdgcn_wmma_f32_16x16x4_f32(
            false, Af, false, Bf, (short)0, accv, false, false);
      }
      if (l < 4) {
        #pragma unroll
        for (int v = 0; v < 8; ++v)
          aO[(v + 8 * hset) * ACTSTR + n0c + col] = fmaxf(accv[v], 0.f);
      } else {
        #pragma unroll
        for (int v = 0; v < 8; ++v) {
          int rn = nwave + v + 8 * hset;
          size_t oidx = ((((size_t)b * NOPS + o) * GG + g) * NN + rn) * 96 + n0c + col;
          out[oidx] = fmaxf(accv[v], 0.f) * msk[v];
        }
      }
    }
    bsel ^= 1;
  }
}

// ============================================================
extern "C" void kernel_launch(void* const* d_in, const int* in_sizes, int n_in,
                              void* d_out, int out_size, void* d_ws, size_t ws_size,
                              hipStream_t stream) {
  (void)in_sizes; (void)n_in; (void)out_size; (void)ws_size;
  const float* pc  = (const float*)d_in[0];
  const float* nrm = (const float*)d_in[1];
  const float* drw = (const float*)d_in[2];
  float* ws  = (float*)d_ws;
  float* out = (float*)d_out;

  static const int cinr[5]  = {51, 32, 48, 64, 80};
  static const int cinp[5]  = {52, 32, 48, 64, 80};
  static const int coutl[5] = {32, 48, 64, 80, 96};
  static const int wstr[5]  = {96, 96, 160, 160, 224};
  static const int owl[5]   = {0, 2496, 4032, 7872, 12992};
  static const int obl[5]   = {0, 32, 80, 144, 224};
  for (int l = 0; l < 5; ++l) {
    const float* W  = (const float*)d_in[3 + 4 * l];
    const float* bb = (const float*)d_in[4 + 4 * l];
    const float* ss = (const float*)d_in[5 + 4 * l];
    const float* oo = (const float*)d_in[6 + 4 * l];
    int sz = cinp[l] * coutl[l];
    prep_layer<<<(sz + 255) / 256, 256, 0, stream>>>(
        W, bb, ss, oo, ws + OFF_WF + owl[l], ws + OFF_BF + obl[l],
        cinr[l], cinp[l], coutl[l], wstr[l]);
  }
  frames_kernel<<<BB * GG, 256, 0, stream>>>(pc, nrm, drw, ws);
  knn_kernel<<<BB * (NN / 128), 128, 0, stream>>>(pc, drw, ws);

  size_t smem = (size_t)(8960 + 96 + 9 + 3 + 3 + 64 + 4 * 2 * ACTSZ) * sizeof(float); // ~86.7 KB
  mlp_kernel<<<BB * NOPS * GG * (NN / 64), 128, smem, stream>>>(
      nrm, drw, ws + OFF_WF, ws + OFF_BF, ws + OFF_FR, ws + OFF_CT,
      ws + OFF_NM, ws + OFF_PN, out);
}